// KV_TaskMemory_70205535420904
// MI455X (gfx1250) — compile-verified
//
#include <hip/hip_runtime.h>
#include <hip/hip_bf16.h>
#include <math.h>

// ---------------------------------------------------------------------------
// KV task-memory read: w = softmax(cos(task, K_m)); rec = sum_m w[m] * V[m]
//
// Phase 1 (tiny, matrix op): dot = K @ task and diag(K @ K^T) via
//   V_WMMA_F32_16X16X4_F32 on a single wave32.
// Phase 2 (340 MB stream, the runtime): async global->LDS staging
//   (global_load_async_to_lds_b128 th:TH_LOAD_NT, ASYNCcnt), ds_load_b128
//   readback, NT b128 stores. Roofline ~14.6 us @ 23.3 TB/s.
// WMMA is deliberately NOT used for the stream: it would reduce per-wave
// contiguous loads from 512 B to 64 B m-strided fragments.
// ---------------------------------------------------------------------------

typedef float f32x4 __attribute__((ext_vector_type(4)));
typedef float v2f   __attribute__((ext_vector_type(2)));
typedef float v8f   __attribute__((ext_vector_type(8)));

#define NSLOT 16          // M memory slots
#define BLK   256         // threads per block (8 waves on wave32)

static __device__ __forceinline__ void async_wait0() {
#if __has_builtin(__builtin_amdgcn_s_wait_asynccnt)
    __builtin_amdgcn_s_wait_asynccnt(0);
#else
    asm volatile("s_wait_asynccnt 0" ::: "memory");
#endif
}

// -------------------- phase 2: streaming weighted reduction ----------------
// out[i] = sum_m w[m] * V[m*nvec4 + i]   (i indexes float4 elements)
__global__ void __launch_bounds__(BLK)
weighted_sum(const float* __restrict__ Vf,
             const float* __restrict__ w,
             float* __restrict__ outf,
             int nvec4) {
    __shared__ f32x4 tile[NSLOT * BLK];   // 64 KB: one 4 KB tile per slot

    const int tid = threadIdx.x;
    const int i   = blockIdx.x * BLK + tid;
    const bool active = (i < nvec4);

    const f32x4* __restrict__ V = (const f32x4*)Vf;

    if (active) {
        // 16 independent async b128 copies per lane, one per memory slot.
        // No dest VGPRs -> low register pressure; NT keeps the 320 MB stream
        // out of the 192 MB L2.
        #pragma unroll
        for (int m = 0; m < NSLOT; ++m) {
            const f32x4* gp = V + (size_t)m * (size_t)nvec4 + i;
            unsigned long long ga = (unsigned long long)(uintptr_t)gp;
            unsigned lo = (unsigned)(uintptr_t)(&tile[m * BLK + tid]);
            asm volatile("global_load_async_to_lds_b128 %0, %1, off th:TH_LOAD_NT"
                         :: "v"(lo), "v"(ga)
                         : "memory");
        }
    }
    async_wait0();
    // Each thread reads back exactly the LDS slots it filled -> no barrier.

    if (active) {
        f32x4 acc = (f32x4)0.0f;
        #pragma unroll
        for (int m = 0; m < NSLOT; ++m) {
            acc += w[m] * tile[m * BLK + tid];
        }
        __builtin_nontemporal_store(acc, (f32x4*)outf + i);
    }
}

// -------------------- phase 1: cosine + softmax weights (WMMA) -------------
// Single wave32. D = A(16x4) x B(4x16) + C with:
//   A[i][kk] = K[i][k0+kk]          (ISA A layout: lane L -> row L&15,
//                                    K cols k0 + 2*(L>=16) + {0,1})
//   B[kk][j] = task[k0+kk]  (bcast) (ISA B layout: lanes 0-15 hold K rows 0-1,
//                                    lanes 16-31 hold K rows 2-3 -> same
//                                    float2 offsets as A)
// => D[i][j]  = dot[i] in every column.
// Second accumulator with B:=A gives Gram matrix; diagonal = ||K_m||^2.
__global__ void __launch_bounds__(32)
compute_weights(const float* __restrict__ task,
                const float* __restrict__ K,
                float* __restrict__ w,
                int key_dim) {
    const int lane = threadIdx.x;          // 0..31, one full wave (EXEC all 1s)
    const int row  = lane & 15;            // A-matrix row fed by this lane
    const int half = (lane >> 4) << 1;     // 0 or 2: K-chunk cols for this half

    v8f cdot = (v8f)0.0f;                  // dot[m] replicated across columns
    v8f cgrm = (v8f)0.0f;                  // Gram matrix K K^T (diag = ||K_m||^2)
    float t2p = 0.0f;                      // partial ||task||^2

    #pragma unroll 4
    for (int k0 = 0; k0 < key_dim; k0 += 4) {
        v2f a = *(const v2f*)(K + row * key_dim + k0 + half);
        v2f b = *(const v2f*)(task + k0 + half);
        cdot = __builtin_amdgcn_wmma_f32_16x16x4_f32(
                   false, a, false, b, (short)0, cdot, false, false);
        cgrm = __builtin_amdgcn_wmma_f32_16x16x4_f32(
                   false, a, false, a, (short)0, cgrm, false, false);
        t2p += b.x * b.x + b.y * b.y;
    }

    // lanes 0..15 hold cols k%4 in {0,1}; lanes 16..31 hold {2,3}
    const float t2 = __shfl(t2p, 0, 32) + __shfl(t2p, 16, 32);

    __shared__ float dotv[NSLOT];
    __shared__ float k2v[NSLOT];

    // C/D layout: lanes 0-15, VGPR r -> (M=r,   N=lane)
    //             lanes 16-31, VGPR r -> (M=8+r, N=lane-16)
    if (lane == 0) {
        #pragma unroll
        for (int r = 0; r < 8; ++r) dotv[r] = cdot[r];
    } else if (lane == 16) {
        #pragma unroll
        for (int r = 0; r < 8; ++r) dotv[8 + r] = cdot[r];
    }
    if (lane < 8) {                        // (M=m, N=m), m = lane < 8
        k2v[lane] = cgrm[lane];
    } else if (lane >= 24) {               // (M=m, N=m), m = lane-16 >= 8
        k2v[lane - 16] = cgrm[lane - 24];
    }
    __syncthreads();                       // single wave: ordering only

    if (lane == 0) {
        const float n1 = sqrtf(t2);
        float cosv[NSLOT];
        float mx = -INFINITY;
        #pragma unroll
        for (int m = 0; m < NSLOT; ++m) {
            cosv[m] = dotv[m] / fmaxf(n1 * sqrtf(k2v[m]), 1e-6f);
            mx = fmaxf(mx, cosv[m]);
        }
        float e[NSLOT];
        float s = 0.0f;
        #pragma unroll
        for (int m = 0; m < NSLOT; ++m) { e[m] = __expf(cosv[m] - mx); s += e[m]; }
        const float inv = 1.0f / s;
        #pragma unroll
        for (int m = 0; m < NSLOT; ++m) w[m] = e[m] * inv;
    }
}

// ---------------------------------------------------------------------------
extern "C" void kernel_launch(void* const* d_in, const int* in_sizes, int n_in,
                              void* d_out, int out_size, void* d_ws, size_t ws_size,
                              hipStream_t stream) {
    const float* task = (const float*)d_in[0];   // [1,1024]
    const float* K    = (const float*)d_in[1];   // [16,1024]
    const float* V0   = (const float*)d_in[2];   // [16,1024,1024]
    const float* V1   = (const float*)d_in[3];   // [16,1024,4096]

    float* w   = (float*)d_ws;                   // 16 softmax weights
    float* out = (float*)d_out;

    const int key_dim = in_sizes[1] / NSLOT;     // 1024
    const int n0      = in_sizes[2] / NSLOT;     // elems per V0 plane (1M)
    const int n1      = in_sizes[3] / NSLOT;     // elems per V1 plane (4M)
    const int n0v4    = n0 / 4;
    const int n1v4    = n1 / 4;

    compute_weights<<<1, 32, 0, stream>>>(task, K, w, key_dim);

    weighted_sum<<<(n0v4 + BLK - 1) / BLK, BLK, 0, stream>>>(V0, w, out, n0v4);
    weighted_sum<<<(n1v4 + BLK - 1) / BLK, BLK, 0, stream>>>(V1, w, out + n0, n1v4);
}